// Reservoir_18640158065252
// MI455X (gfx1250) — compile-verified
//
#include <hip/hip_runtime.h>
#include <hip/hip_bf16.h>
#include <math.h>

// Problem constants (from reference)
#define T_STEPS 4096
#define IN_DIM  128
#define HID     2048
#define LEAK    0.9f

// Recurrence kernel config
#define NB      64               // persistent blocks (must all be co-resident)
#define ROWS    (HID / NB)       // 32 hidden rows per block
#define WSTRIDE 2052             // 2048 + 4 pad floats -> conflict-free LDS banks
#define SCAN_THREADS 256
// dynamic LDS: w tile + act vector + wave partials
#define SCAN_LDS_FLOATS (ROWS * WSTRIDE + HID + (SCAN_THREADS / 32) * 32)
#define SCAN_LDS_BYTES  (SCAN_LDS_FLOATS * 4)

typedef __attribute__((ext_vector_type(2))) float v2f;
typedef __attribute__((ext_vector_type(8))) float v8f;

// ---------------------------------------------------------------------------
// CDNA5 async global->LDS copy (ASYNCcnt path, bypasses VGPRs).
// LDS byte address = low 32 bits of the generic pointer.
// ---------------------------------------------------------------------------
__device__ __forceinline__ void async_g2l_b128(const float* lds_dst, const float* gsrc) {
  unsigned lds_off = (unsigned)(size_t)lds_dst;
  asm volatile("global_load_async_to_lds_b128 %0, %1, off"
               :
               : "v"(lds_off), "v"(gsrc)
               : "memory");
}
__device__ __forceinline__ void wait_async0() {
  asm volatile("s_wait_asynccnt 0x0" ::: "memory");
}

// ---------------------------------------------------------------------------
// Kernel 1: input projection  u[t][h] = sum_k input[t][k] * w_ih[h][k] + b_ih[h]
// Full-precision f32 WMMA (V_WMMA_F32_16X16X4_F32), one 16x16 tile per wave.
// ---------------------------------------------------------------------------
__global__ __launch_bounds__(256) void inproj_wmma(
    const float* __restrict__ inp,   // [T, IN_DIM]
    const float* __restrict__ w_ih,  // [HID, IN_DIM]
    const float* __restrict__ b_ih,  // [HID]
    float* __restrict__ u)           // [T, HID]  (all_hid region of d_out)
{
  const int wave = threadIdx.x >> 5;
  const int lane = threadIdx.x & 31;
  const int tile = blockIdx.x * 8 + wave;      // T/16 * HID/16 = 256*128 tiles
  const int tT = tile >> 7;                    // tile row over T
  const int tH = tile & 127;                   // tile col over HID
  const int half = lane >> 4;                  // 0: K0/K1, 1: K2/K3 (ISA layout)
  const int rc = lane & 15;                    // A row / B col / C col

  // A fragment source: input rows (row-major over k) -> contiguous K pair
  const float* aBase = inp + ((size_t)(tT * 16 + rc)) * IN_DIM + half * 2;
  // B fragment: B[k][n] = w_ih[n][k]; w_ih row-major over k -> contiguous pair
  const float* bBase = w_ih + ((size_t)(tH * 16 + rc)) * IN_DIM + half * 2;

  v8f c = {0.f, 0.f, 0.f, 0.f, 0.f, 0.f, 0.f, 0.f};
#pragma unroll
  for (int k = 0; k < IN_DIM; k += 4) {
    v2f a = { aBase[k], aBase[k + 1] };
    v2f b = { bBase[k], bBase[k + 1] };
    c = __builtin_amdgcn_wmma_f32_16x16x4_f32(
        /*neg_a=*/false, a, /*neg_b=*/false, b,
        /*c_mod=*/(short)0, c, /*reuse_a=*/false, /*reuse_b=*/false);
  }

  const float bias = b_ih[tH * 16 + rc];
#pragma unroll
  for (int v = 0; v < 8; ++v) {
    const int row = v + half * 8;              // C layout: lane>=16 holds rows 8..15
    u[((size_t)(tT * 16 + row)) * HID + tH * 16 + rc] = c[v] + bias;
  }
}

// ---------------------------------------------------------------------------
// Device-wide barrier: monotonic counter, agent-scope release/acquire.
// ---------------------------------------------------------------------------
__device__ __forceinline__ void grid_barrier(unsigned* cnt, unsigned target) {
  __syncthreads();
  if (threadIdx.x == 0) {
    __threadfence();  // release all prior global writes to device scope
    __hip_atomic_fetch_add(cnt, 1u, __ATOMIC_RELEASE, __HIP_MEMORY_SCOPE_AGENT);
    while (__hip_atomic_load(cnt, __ATOMIC_ACQUIRE, __HIP_MEMORY_SCOPE_AGENT) < target) {
      __builtin_amdgcn_s_sleep(2);
    }
    __threadfence();  // acquire: invalidate stale lines before re-reading act
  }
  __syncthreads();
}

// ---------------------------------------------------------------------------
// Kernel 2: persistent sequential scan. Each of the 64 blocks owns 32 hidden
// rows and pins its 32x2048 slice of w_hh in LDS (262KB of the 320KB/WGP),
// staged via the CDNA5 async global->LDS DMA path. Per step: async-stage
// act_{t-1} (8KB) into LDS, 8 waves split K (lane = row), conflict-free b128
// LDS streaming, wave0 finalizes leak+tanh, device-wide barrier.
// ---------------------------------------------------------------------------
__global__ __launch_bounds__(SCAN_THREADS) void reservoir_scan(
    const float* __restrict__ w_hh,   // [HID, HID]
    float* __restrict__ actOut,       // d_out          [T, HID] : all_act
    float* __restrict__ hidOut,       // d_out + T*HID  [T, HID] : pre-filled with u, becomes all_hid
    unsigned* __restrict__ barrier_cnt)
{
  extern __shared__ float smem[];
  float* w_s   = smem;                    // ROWS * WSTRIDE
  float* act_s = smem + ROWS * WSTRIDE;   // HID
  float* part  = act_s + HID;             // 8 waves * 32 lanes

  const int tid  = threadIdx.x;
  const int wave = tid >> 5;
  const int lane = tid & 31;
  const int rowBase = blockIdx.x * ROWS;

  // Stage this block's 32 rows of w_hh into LDS once via async b128 DMA
  // (row stride padded; 4-float chunks never cross a row boundary).
  for (int i = tid; i < ROWS * HID / 4; i += SCAN_THREADS) {
    const int f = i * 4;
    const int r = f >> 11;                // / HID
    const int k = f & (HID - 1);
    async_g2l_b128(w_s + r * WSTRIDE + k,
                   w_hh + (size_t)(rowBase + r) * HID + k);
  }
  wait_async0();
  __syncthreads();

  float nh_prev = 0.0f;                   // hidden-state carry for row (rowBase+tid), tid<32
  const int kb = wave * (HID / 8);        // this wave's K chunk (256 wide)

  for (int t = 0; t < T_STEPS; ++t) {
    // stage act_{t-1} (zeros at t==0, matching the reference init carry)
    if (t == 0) {
      for (int idx = tid; idx < HID; idx += SCAN_THREADS) act_s[idx] = 0.0f;
    } else {
      const float* ap = actOut + (size_t)(t - 1) * HID;
      const int idx = tid * 8;            // 256 threads x 8 floats = 2048
      async_g2l_b128(act_s + idx,     ap + idx);
      async_g2l_b128(act_s + idx + 4, ap + idx + 4);
      wait_async0();
    }
    __syncthreads();

    // lane l accumulates row l over this wave's K chunk; b128 LDS streams
    float acc = 0.0f;
    const float* wr = w_s + lane * WSTRIDE + kb;
    const float* av = act_s + kb;
#pragma unroll 4
    for (int j = 0; j < HID / 8; j += 4) {
      const float4 w4 = *reinterpret_cast<const float4*>(wr + j);
      const float4 a4 = *reinterpret_cast<const float4*>(av + j);
      acc = fmaf(w4.x, a4.x, acc);
      acc = fmaf(w4.y, a4.y, acc);
      acc = fmaf(w4.z, a4.z, acc);
      acc = fmaf(w4.w, a4.w, acc);
    }
    part[wave * 32 + lane] = acc;
    __syncthreads();

    // wave 0 finalizes: sum 8 partials, leak, tanh, write both outputs
    if (tid < 32) {
      float s = part[tid];
#pragma unroll
      for (int w = 1; w < SCAN_THREADS / 32; ++w) s += part[w * 32 + tid];
      const size_t o = (size_t)t * HID + rowBase + tid;
      const float u = hidOut[o];                       // precomputed inp_proj
      const float nh = (1.0f - LEAK) * nh_prev + LEAK * (u + s);
      nh_prev = nh;
      hidOut[o] = nh;                                  // all_hid[t]
      actOut[o] = tanhf(nh);                           // all_act[t]
    }

    grid_barrier(barrier_cnt, (unsigned)(t + 1) * NB);
  }
}

// ---------------------------------------------------------------------------
extern "C" void kernel_launch(void* const* d_in, const int* in_sizes, int n_in,
                              void* d_out, int out_size, void* d_ws, size_t ws_size,
                              hipStream_t stream) {
  (void)in_sizes; (void)n_in; (void)out_size; (void)ws_size;
  const float* inp  = (const float*)d_in[0];  // [T, IN_DIM]
  const float* w_ih = (const float*)d_in[1];  // [HID, IN_DIM]
  const float* b_ih = (const float*)d_in[2];  // [HID]
  const float* w_hh = (const float*)d_in[3];  // [HID, HID]

  float* actOut = (float*)d_out;                       // all_act [T, HID]
  float* hidOut = actOut + (size_t)T_STEPS * HID;      // all_hid [T, HID]
  unsigned* cnt = (unsigned*)d_ws;

  // zero the barrier counter each launch (async memset is graph-capturable)
  hipMemsetAsync(d_ws, 0, sizeof(unsigned), stream);

  // allow >64KB dynamic LDS for the persistent scan kernel (idempotent)
  hipFuncSetAttribute(reinterpret_cast<const void*>(reservoir_scan),
                      hipFuncAttributeMaxDynamicSharedMemorySize, SCAN_LDS_BYTES);

  // Phase 1: u = input @ w_ih.T + b_ih  -> stored into the all_hid region
  inproj_wmma<<<(T_STEPS / 16) * (HID / 16) / 8, 256, 0, stream>>>(inp, w_ih, b_ih, hidOut);

  // Phase 2: persistent sequential recurrence with device-wide barriers
  reservoir_scan<<<NB, SCAN_THREADS, SCAN_LDS_BYTES, stream>>>(w_hh, actOut, hidOut, cnt);
}